// RotationAttention_44839458570375
// MI455X (gfx1250) — compile-verified
//
#include <hip/hip_runtime.h>
#include <math.h>

// Problem constants (B, M, D, H) = (32, 64, 512, 512)
#define B_ 32
#define M_ 64
#define D_ 512
#define H_ 512
#define TEMP_INV (1.0f/0.3f)
#define EPS_ 1e-12f

typedef float v2f __attribute__((ext_vector_type(2)));
typedef float v8f __attribute__((ext_vector_type(8)));

// LDS layout (floats). Rows of the 64x512 refer block are padded to 516
// floats so WMMA A/B fragment loads (16 rows at same column) spread across
// all 64 LDS banks instead of hitting one bank 16-way.
#define RS_STRIDE 516
#define RS_OFF    0
#define IM_OFF    (64*RS_STRIDE)
#define SS_OFF    (IM_OFF + D_)
#define RR_OFF    (SS_OFF + M_*M_)
#define RI_OFF    (RR_OFF + M_*9)
#define SMEM_FLOATS (RI_OFF + M_*9)

// ---------------------------------------------------------------------------
// Kernel 1 (per batch): S = Q * refer^T via V_WMMA_F32_16X16X4_F32, then
// factorized softmax + weighted 3x3 rotation sums -> out6 (B*M x 6).
// ---------------------------------------------------------------------------
__global__ void __launch_bounds__(256)
rotattn_scores_out6(const float* __restrict__ img,
                    const float* __restrict__ refer,
                    const float* __restrict__ rr_g,
                    float* __restrict__ out6)
{
  extern __shared__ float smem[];
  float* Rs = smem + RS_OFF;   // refer[b] 64 x 512, padded rows
  float* Im = smem + IM_OFF;   // img[b]   512
  float* Ss = smem + SS_OFF;   // S        64 x 64
  float* Rr = smem + RR_OFF;   // refer_r[b] 64 x 9
  float* Ri = smem + RI_OFF;   // inv(refer_r[b]) 64 x 9

  const int b    = blockIdx.x;
  const int tid  = threadIdx.x;
  const int lane = tid & 31;
  const int wave = tid >> 5;

  // ---- Phase 0: stage operands into LDS ----
  for (int k = tid; k < D_; k += 256) Im[k] = img[b*D_ + k];
  const float* refb = refer + (size_t)b * M_ * D_;
  for (int idx = tid * 4; idx < M_ * D_; idx += 256 * 4) {
    const int row = idx >> 9;         // / 512
    const int col = idx & (D_ - 1);
    const float4 v = *(const float4*)(refb + idx);
    *(float4*)(Rs + row * RS_STRIDE + col) = v;
  }
  for (int i = tid; i < M_*9; i += 256) Rr[i] = rr_g[(size_t)b*M_*9 + i];
  __syncthreads();

  // ---- Phase 1: 16 output tiles (4x4 grid of 16x16); 2 tiles per wave ----
  // A fragment (16x4 f32): lane holds {A[m][kb], A[m][kb+1]}, m=lane%16,
  // kb = 2*(lane/16).  B fragment (4x16): lane holds {B[kb][n], B[kb+1][n]}.
  const int mlane = lane & 15;
  const int koff  = (lane >> 4) << 1;          // 0 or 2
  for (int t = wave; t < 16; t += 8) {
    const int ti = t >> 2, tj = t & 3;
    const float* arow = Rs + (ti*16 + mlane) * RS_STRIDE;  // Q row source
    const float* brow = Rs + (tj*16 + mlane) * RS_STRIDE;  // K row source
    v8f acc = {};
    for (int kk = 0; kk < D_; kk += 4) {
      const int k = kk + koff;
      const v2f iv = *(const v2f*)(Im + k);
      const v2f rv = *(const v2f*)(arow + k);
      v2f qa; qa.x = iv.x - rv.x; qa.y = iv.y - rv.y;   // Q = img - refer
      const v2f bv = *(const v2f*)(brow + k);
      acc = __builtin_amdgcn_wmma_f32_16x16x4_f32(false, qa, false, bv,
                                                  (short)0, acc, false, false);
    }
    // C layout: VGPR v -> M = v + 8*(lane/16), N = lane%16
    const int mbase = ti*16 + ((lane >> 4) << 3);
    const int ncol  = tj*16 + mlane;
    #pragma unroll
    for (int v = 0; v < 8; ++v)
      Ss[(mbase + v) * 64 + ncol] = acc[v];
  }
  __syncthreads();

  // ---- Phase 2a: 3x3 inverse of refer_r via adjugate ----
  if (tid < M_) {
    const float* r = Rr + tid*9;
    const float a=r[0], bb=r[1], c=r[2], d=r[3], e=r[4], f=r[5],
                g=r[6], h=r[7], i=r[8];
    const float A0 = e*i - f*h, A1 = -(d*i - f*g), A2 = d*h - e*g;
    const float inv = 1.0f / (a*A0 + bb*A1 + c*A2);
    float* o = Ri + tid*9;
    o[0]=A0*inv;  o[1]=-(bb*i - c*h)*inv;  o[2]= (bb*f - c*e)*inv;
    o[3]=A1*inv;  o[4]= (a*i  - c*g)*inv;  o[5]=-(a*f  - c*d)*inv;
    o[6]=A2*inv;  o[7]=-(a*h  - bb*g)*inv; o[8]= (a*e  - bb*d)*inv;
  }
  __syncthreads();

  // ---- Phase 2b: factorized softmax + weighted rotation sums ----
  // w[j1,j2] = a[j1] b[j2] / (sum_a * sum_b);  out9 = Mb @ Ma.
  if (tid < M_) {
    const float* srow = Ss + tid*64;
    float mx = -INFINITY, mn = INFINITY;
    for (int j = 0; j < M_; ++j) {
      const float s = srow[j] * TEMP_INV;
      mx = fmaxf(mx, s); mn = fminf(mn, s);
    }
    float Ma[9] = {}, Mb[9] = {};
    float sa = 0.f, sb = 0.f;
    for (int j = 0; j < M_; ++j) {
      const float s  = srow[j] * TEMP_INV;
      const float aw = expf(s - mx);
      const float bw = expf(mn - s);   // exp(-s - max(-s))
      sa += aw; sb += bw;
      const float* rj = Rr + j*9;
      const float* qj = Ri + j*9;
      #pragma unroll
      for (int u = 0; u < 9; ++u) { Ma[u] += aw*rj[u]; Mb[u] += bw*qj[u]; }
    }
    const float invAB = 1.0f / (sa * sb);
    float* o = out6 + ((size_t)b*M_ + tid)*6;
    #pragma unroll
    for (int p = 0; p < 2; ++p)
      #pragma unroll
      for (int q = 0; q < 3; ++q) {
        float acc2 = 0.f;
        #pragma unroll
        for (int u = 0; u < 3; ++u) acc2 += Mb[p*3+u]*Ma[u*3+q];
        o[p*3+q] = acc2 * invAB;
      }
  }
}

// ---------------------------------------------------------------------------
// Kernel 2: W_eff = W_enc @ W1 (6x512), b_eff = b_enc @ W1 + b1.
// ---------------------------------------------------------------------------
__global__ void __launch_bounds__(256)
rotattn_weff(const float* __restrict__ W_enc, const float* __restrict__ b_enc,
             const float* __restrict__ W1, const float* __restrict__ b1,
             float* __restrict__ W_eff, float* __restrict__ b_eff)
{
  const int h = blockIdx.x*256 + threadIdx.x;
  if (h >= H_) return;
  float acc[6] = {};
  float bacc = 0.f;
  for (int d = 0; d < D_; ++d) {
    const float w1v = W1[d*H_ + h];
    #pragma unroll
    for (int p = 0; p < 6; ++p) acc[p] += W_enc[p*D_ + d] * w1v;
    bacc += b_enc[d] * w1v;
  }
  #pragma unroll
  for (int p = 0; p < 6; ++p) W_eff[p*H_ + h] = acc[p];
  b_eff[h] = bacc + b1[h];
}

// ---------------------------------------------------------------------------
// Kernel 3: per (b,i): hidden = relu(out6 @ W_eff + b_eff) folded into
// r6 = hidden @ W2 + b2, then Gram-Schmidt recovery and refer_r @ R.
// ---------------------------------------------------------------------------
__global__ void __launch_bounds__(256)
rotattn_head(const float* __restrict__ out6,
             const float* __restrict__ W_eff, const float* __restrict__ b_eff,
             const float* __restrict__ W2, const float* __restrict__ b2,
             const float* __restrict__ rr_g, float* __restrict__ outR)
{
  const int idx = blockIdx.x*256 + threadIdx.x;   // b*64 + i, 0..2047
  const float* o6 = out6 + (size_t)idx*6;
  const float x0=o6[0], x1=o6[1], x2=o6[2], x3=o6[3], x4=o6[4], x5=o6[5];
  float acc[6] = {};
  for (int h = 0; h < H_; ++h) {
    float hid = b_eff[h]
      + x0*W_eff[h]      + x1*W_eff[H_+h]   + x2*W_eff[2*H_+h]
      + x3*W_eff[3*H_+h] + x4*W_eff[4*H_+h] + x5*W_eff[5*H_+h];
    hid = fmaxf(hid, 0.0f);
    const float* w2 = W2 + h*6;
    #pragma unroll
    for (int c = 0; c < 6; ++c) acc[c] += hid * w2[c];
  }
  float r1[3], r2[3], r3[3];
  #pragma unroll
  for (int c = 0; c < 3; ++c) { r1[c] = acc[c] + b2[c]; r2[c] = acc[c+3] + b2[c+3]; }
  const float dot = r1[0]*r2[0] + r1[1]*r2[1] + r1[2]*r2[2];
  #pragma unroll
  for (int c = 0; c < 3; ++c) r2[c] -= dot * r1[c];
  r3[0] = r1[1]*r2[2] - r1[2]*r2[1];
  r3[1] = r1[2]*r2[0] - r1[0]*r2[2];
  r3[2] = r1[0]*r2[1] - r1[1]*r2[0];
  auto norm3 = [](float* v) {
    const float n = sqrtf(v[0]*v[0] + v[1]*v[1] + v[2]*v[2]);
    const float s = 1.0f / fmaxf(n, EPS_);
    v[0] *= s; v[1] *= s; v[2] *= s;
  };
  norm3(r1); norm3(r2); norm3(r3);
  // R columns are r1,r2,r3 (stack axis=-1); F = refer_r[b,i] @ R
  const float* rr = rr_g + (size_t)idx*9;
  float* o = outR + (size_t)idx*9;
  #pragma unroll
  for (int p = 0; p < 3; ++p) {
    const float a0 = rr[p*3+0], a1 = rr[p*3+1], a2 = rr[p*3+2];
    o[p*3+0] = a0*r1[0] + a1*r1[1] + a2*r1[2];
    o[p*3+1] = a0*r2[0] + a1*r2[1] + a2*r2[2];
    o[p*3+2] = a0*r3[0] + a1*r3[1] + a2*r3[2];
  }
}

// ---------------------------------------------------------------------------
extern "C" void kernel_launch(void* const* d_in, const int* in_sizes, int n_in,
                              void* d_out, int out_size, void* d_ws, size_t ws_size,
                              hipStream_t stream)
{
  (void)in_sizes; (void)n_in; (void)out_size; (void)ws_size;
  const float* img   = (const float*)d_in[0];
  const float* refer = (const float*)d_in[1];
  const float* rr    = (const float*)d_in[2];
  const float* W_enc = (const float*)d_in[3];
  const float* b_enc = (const float*)d_in[4];
  const float* W1    = (const float*)d_in[5];
  const float* b1    = (const float*)d_in[6];
  const float* W2    = (const float*)d_in[7];
  const float* b2    = (const float*)d_in[8];
  float* out = (float*)d_out;

  float* ws    = (float*)d_ws;
  float* out6  = ws;                     // 2048 * 6
  float* W_eff = ws + B_*M_*6;           // 6 * 512
  float* b_eff = W_eff + 6*H_;           // 512

  const size_t smem = SMEM_FLOATS * sizeof(float);   // ~152 KB (< 320 KB/WGP)
  rotattn_scores_out6<<<B_, 256, smem, stream>>>(img, refer, rr, out6);
  rotattn_weff<<<2, 256, 0, stream>>>(W_enc, b_enc, W1, b1, W_eff, b_eff);
  rotattn_head<<<8, 256, 0, stream>>>(out6, W_eff, b_eff, W2, b2, rr, out);
}